// GCN_50345606643998
// MI455X (gfx1250) — compile-verified
//
#include <hip/hip_runtime.h>

// ---------------------------------------------------------------------------
// GCN 2-layer forward for MI455X (gfx1250, wave32).
//   agg = scatter_add(h[src] -> dst)   : L2-resident atomics (51MB tables << 192MB L2)
//   h'  = relu(agg @ W + b)            : V_WMMA_F32_16X16X4_F32 (fp32 matrix pipe)
// ---------------------------------------------------------------------------

#define N_NODES   100000
#define N_EDGES   1600000
#define IN_FEATS  128
#define N_HIDDEN  128
#define N_CLASSES 64

typedef __attribute__((ext_vector_type(2))) float v2f;
typedef __attribute__((ext_vector_type(8))) float v8f;

// ---------------------------------------------------------------------------
// Zero-fill (float4 stores, grid-stride)
// ---------------------------------------------------------------------------
__global__ void gcn_zero_f32(float4* __restrict__ p, int n4) {
    int i      = blockIdx.x * blockDim.x + threadIdx.x;
    int stride = gridDim.x * blockDim.x;
    for (; i < n4; i += stride)
        p[i] = make_float4(0.f, 0.f, 0.f, 0.f);
}

// ---------------------------------------------------------------------------
// Edge-parallel scatter-add: one wave32 per edge, lane handles 4 floats.
// 128 feats = 32 lanes * float4. Gather hits L2 (feature table is 51.2 MB).
// unsafeAtomicAdd -> global_atomic_add_f32 (L2 RMW unit, no CAS loop).
// ---------------------------------------------------------------------------
__global__ void gcn_scatter_add_128(const float* __restrict__ h,
                                    const int*   __restrict__ src,
                                    const int*   __restrict__ dst,
                                    float*       __restrict__ agg,
                                    int n_edges) {
    const int wave = threadIdx.x >> 5;
    const int lane = threadIdx.x & 31;
    const int wpb  = blockDim.x >> 5;
    const int e    = blockIdx.x * wpb + wave;
    if (e >= n_edges) return;

    const int s = src[e];
    const int d = dst[e];

    const float4 v = reinterpret_cast<const float4*>(h + (long)s * 128)[lane];
    float* out = agg + (long)d * 128 + lane * 4;
    unsafeAtomicAdd(out + 0, v.x);
    unsafeAtomicAdd(out + 1, v.y);
    unsafeAtomicAdd(out + 2, v.z);
    unsafeAtomicAdd(out + 3, v.w);
}

// ---------------------------------------------------------------------------
// Fused GEMM + bias + ReLU using V_WMMA_F32_16X16X4_F32.
//   A: [n_rows, K] row-major (agg), W: [K, NC] row-major, out: [n_rows, NC].
// One wave computes a 16-row stripe across all NC columns (NC/16 acc tiles),
// so the A fragment is loaded once per k-step and reused for every col tile.
//
// VGPR layouts per CDNA5 ISA 7.12.2 (wave32):
//   A 16x4 f32 : lane M = lane&15 ; VGPRs hold K = {2*half, 2*half+1}, half=lane>>4
//   B 4x16 f32 : lane N = lane&15 ; VGPRs hold K = {2*half, 2*half+1}
//   C/D 16x16  : VGPR i -> M = i + 8*half, N = lane&15
// ---------------------------------------------------------------------------
template <int K, int NC>
__global__ void gcn_gemm_bias_relu(const float* __restrict__ A,
                                   const float* __restrict__ W,
                                   const float* __restrict__ bias,
                                   float*       __restrict__ out,
                                   int n_rows) {
    constexpr int NT = NC / 16;

    const int wave = threadIdx.x >> 5;
    const int lane = threadIdx.x & 31;
    const int wpb  = blockDim.x >> 5;
    const int tile = blockIdx.x * wpb + wave;   // uniform per wave -> EXEC stays all-1s
    const int row0 = tile * 16;
    if (row0 >= n_rows) return;

    const int half = lane >> 4;   // 0 or 1
    const int l15  = lane & 15;

    v8f acc[NT];
#pragma unroll
    for (int t = 0; t < NT; ++t) acc[t] = 0.0f;

    const float* Arow = A + (long)(row0 + l15) * K;

    for (int k0 = 0; k0 < K; k0 += 4) {
        const int kk = k0 + 2 * half;
        v2f a;
        a.x = Arow[kk];
        a.y = Arow[kk + 1];
#pragma unroll
        for (int t = 0; t < NT; ++t) {
            const int col = t * 16 + l15;
            v2f b;
            b.x = W[(long)kk * NC + col];
            b.y = W[(long)(kk + 1) * NC + col];
            acc[t] = __builtin_amdgcn_wmma_f32_16x16x4_f32(
                /*neg_a=*/false, a, /*neg_b=*/false, b,
                /*c_mod=*/(short)0, acc[t],
                /*reuse_a=*/false, /*reuse_b=*/false);
        }
    }

#pragma unroll
    for (int t = 0; t < NT; ++t) {
        const int col = t * 16 + l15;
        const float bv = bias[col];
#pragma unroll
        for (int i = 0; i < 8; ++i) {
            const int r = row0 + i + 8 * half;
            float v = acc[t][i] + bv;
            v = v > 0.f ? v : 0.f;
            out[(long)r * NC + col] = v;
        }
    }
}

// ---------------------------------------------------------------------------
// Launcher
// inputs: 0=features[N,128] 1=W1[128,128] 2=b1[128] 3=W2[128,64] 4=b2[64]
//         5=src[E] 6=dst[E]        output: [N,64] f32
// ---------------------------------------------------------------------------
extern "C" void kernel_launch(void* const* d_in, const int* in_sizes, int n_in,
                              void* d_out, int out_size, void* d_ws, size_t ws_size,
                              hipStream_t stream) {
    const float* features = (const float*)d_in[0];
    const float* W1       = (const float*)d_in[1];
    const float* b1       = (const float*)d_in[2];
    const float* W2       = (const float*)d_in[3];
    const float* b2       = (const float*)d_in[4];
    const int*   src      = (const int*)d_in[5];
    const int*   dst      = (const int*)d_in[6];
    float*       outp     = (float*)d_out;

    float* buf0 = (float*)d_ws;                          // agg   [N,128] 51.2 MB
    float* buf1 = buf0 + (long)N_NODES * N_HIDDEN;       // h1    [N,128] 51.2 MB

    const int n4        = N_NODES * N_HIDDEN / 4;        // float4 count
    const int zeroBlks  = 4096;
    const int edgeBlks  = (N_EDGES + 7) / 8;             // 8 waves/block, 1 edge/wave
    const int rowTiles  = (N_NODES + 15) / 16;           // 6250
    const int gemmBlks  = (rowTiles + 7) / 8;            // 8 waves/block

    // ---- layer 1 ----
    gcn_zero_f32<<<zeroBlks, 256, 0, stream>>>((float4*)buf0, n4);
    gcn_scatter_add_128<<<edgeBlks, 256, 0, stream>>>(features, src, dst, buf0, N_EDGES);
    gcn_gemm_bias_relu<IN_FEATS, N_HIDDEN>
        <<<gemmBlks, 256, 0, stream>>>(buf0, W1, b1, buf1, N_NODES);

    // ---- layer 2 ----
    gcn_zero_f32<<<zeroBlks, 256, 0, stream>>>((float4*)buf0, n4);
    gcn_scatter_add_128<<<edgeBlks, 256, 0, stream>>>(buf1, src, dst, buf0, N_EDGES);
    gcn_gemm_bias_relu<N_HIDDEN, N_CLASSES>
        <<<gemmBlks, 256, 0, stream>>>(buf0, W2, b2, outp, N_NODES);
}